// GlobalRpeDecompCrossAttention_80255758893720
// MI455X (gfx1250) — compile-verified
//
#include <hip/hip_runtime.h>
#include <math.h>

typedef __attribute__((ext_vector_type(2))) float v2f;
typedef __attribute__((ext_vector_type(8))) float v8f;

#define DIMC   256
#define NHEADS 8
#define HD     32
#define QLEN   900
#define BATCH  2
#define KVLEN  4096
#define RPEH   512
#define GRIDW  64
#define STRIDEV 16

__device__ __forceinline__ v8f wmma4(v2f a, v2f b, v8f c) {
    // D(16x16,f32) = A(16x4,f32) x B(4x16,f32) + C
    return __builtin_amdgcn_wmma_f32_16x16x4_f32(
        /*neg_a=*/false, a, /*neg_b=*/false, b,
        /*c_mod=*/(short)0, c, /*reuse_a=*/false, /*reuse_b=*/false);
}

// ---------------------------------------------------------------------------
// Projection GEMM: C[M x 256] = (A[M x 256] @ W^T + bias) * scale
// W is [256(out) x 256(in)] row-major, so B[k][n] = W[n][k].
// One wave computes a 16x64 output slab: one A fragment feeds 4 WMMAs
// (4x A-reuse), 256 x V_WMMA_F32_16X16X4_F32 per wave.
// Out-of-range rows are CLAMPED (not masked): they compute garbage that is
// discarded at the store, keeping the inner loop free of EXEC manipulation.
// ---------------------------------------------------------------------------
__global__ __launch_bounds__(32)
void proj_gemm64(const float* __restrict__ A, const float* __restrict__ W,
                 const float* __restrict__ bias, float* __restrict__ C,
                 int M, float scale) {
    const int lane = threadIdx.x & 31;
    const int r    = lane & 15;
    const int hi   = lane >> 4;
    const int kb0  = hi * 2;
    const int m0   = blockIdx.x * 16;
    const int n0   = blockIdx.y * 64;

    const int arow   = m0 + r;
    const int arow_c = (arow < M) ? arow : (M - 1);     // clamp, no divergence
    const float* ap  = A + (size_t)arow_c * DIMC + kb0;
    const float* wp0 = W + (size_t)(n0 + r) * DIMC + kb0;
    const float* wp1 = wp0 + (size_t)16 * DIMC;
    const float* wp2 = wp0 + (size_t)32 * DIMC;
    const float* wp3 = wp0 + (size_t)48 * DIMC;

    v8f acc[4] = {};
#pragma unroll 4
    for (int k = 0; k < DIMC; k += 4) {
        const v2f a = *(const v2f*)(ap + k);
        acc[0] = wmma4(a, *(const v2f*)(wp0 + k), acc[0]);
        acc[1] = wmma4(a, *(const v2f*)(wp1 + k), acc[1]);
        acc[2] = wmma4(a, *(const v2f*)(wp2 + k), acc[2]);
        acc[3] = wmma4(a, *(const v2f*)(wp3 + k), acc[3]);
    }

    const int rbase = m0 + (hi << 3);
#pragma unroll
    for (int t = 0; t < 4; ++t) {
        const int   col = n0 + t * 16 + r;
        const float bv  = bias[col];
#pragma unroll
        for (int v = 0; v < 8; ++v) {
            const int row = rbase + v;
            if (row < M) C[(size_t)row * DIMC + col] = (acc[t][v] + bv) * scale;
        }
    }
}

// ---------------------------------------------------------------------------
// RPE tables: rpex[b,q,x,h], rpey[b,q,y,h]  (2 -> 512 relu -> 8 MLPs)
// One block per (b,q); threads 0..63 handle x-positions, 64..127 y-positions.
// ---------------------------------------------------------------------------
__global__ __launch_bounds__(128)
void rpe_kernel(const float* __restrict__ refp,
                const float* __restrict__ w1x, const float* __restrict__ b1x,
                const float* __restrict__ w2x,
                const float* __restrict__ w1y, const float* __restrict__ b1y,
                const float* __restrict__ w2y,
                float* __restrict__ rpex, float* __restrict__ rpey) {
    const int bq = blockIdx.x;
    const int t  = threadIdx.x;
    const float cx = refp[bq * 4 + 0], cy = refp[bq * 4 + 1];
    const float sw = refp[bq * 4 + 2], sh = refp[bq * 4 + 3];
    const float mult = (float)(GRIDW * STRIDEV);          // 64*16 = 1024
    const float x0 = (cx - sw * 0.5f) * mult, x1 = (cx + sw * 0.5f) * mult;
    const float y0 = (cy - sh * 0.5f) * mult, y1 = (cy + sh * 0.5f) * mult;

    const int  isY = t >= GRIDW;
    const int  pos = t & (GRIDW - 1);
    const float pc = ((float)pos + 0.5f) * (float)STRIDEV;
    const float d0 = (isY ? y0 : x0) - pc;
    const float d1 = (isY ? y1 : x1) - pc;
    const float* w1 = isY ? w1y : w1x;
    const float* b1 = isY ? b1y : b1x;
    const float* w2 = isY ? w2y : w2x;

    float acc[NHEADS];
#pragma unroll
    for (int h = 0; h < NHEADS; ++h) acc[h] = 0.f;
    for (int rr = 0; rr < RPEH; ++rr) {
        float tmp = fmaxf(fmaf(w1[rr * 2], d0, fmaf(w1[rr * 2 + 1], d1, b1[rr])), 0.f);
#pragma unroll
        for (int h = 0; h < NHEADS; ++h)
            acc[h] = fmaf(tmp, w2[h * RPEH + rr], acc[h]);
    }
    float* dst = (isY ? rpey : rpex) + ((size_t)bq * GRIDW + pos) * NHEADS;
#pragma unroll
    for (int h = 0; h < NHEADS; ++h) dst[h] = acc[h];
}

// ---------------------------------------------------------------------------
// Flash attention: one wave per (16-query tile, head, batch).
// scores = qhat·khat^T + rpex + rpey + mask*-100 ; online softmax ; P@V.
// All matmuls are V_WMMA_F32_16X16X4_F32. Out-of-range query rows are
// clamped (duplicate row 899); their results are discarded at the store.
// ---------------------------------------------------------------------------
__global__ __launch_bounds__(32)
void attn_kernel(const float* __restrict__ qhat, const float* __restrict__ khat,
                 const float* __restrict__ vhat,
                 const float* __restrict__ rpex, const float* __restrict__ rpey,
                 const int* __restrict__ amask, float* __restrict__ ctx) {
    __shared__ float ldsP[16 * 16];

    const int lane = threadIdx.x & 31;
    const int r    = lane & 15;
    const int hi   = lane >> 4;
    const int kb0  = hi * 2;
    const int q0   = blockIdx.x * 16;
    const int h    = blockIdx.y;
    const int b    = blockIdx.z;

    // Preload Q A-fragments: 16x32 tile = 8 fragments of 16x4 (row clamped)
    v2f aq[8];
    {
        const int qrow = q0 + r;
        const int qrc  = (qrow < QLEN) ? qrow : (QLEN - 1);
        const float* qp = qhat + ((size_t)(b * QLEN + qrc)) * DIMC + h * HD + kb0;
#pragma unroll
        for (int kk = 0; kk < 8; ++kk) aq[kk] = *(const v2f*)(qp + kk * 4);
    }

    // rpe row pointers for the 8 (clamped) rows this lane touches
    size_t rpebase[8];
#pragma unroll
    for (int v = 0; v < 8; ++v) {
        int qrow = q0 + (hi << 3) + v;
        if (qrow >= QLEN) qrow = QLEN - 1;
        rpebase[v] = (size_t)(b * QLEN + qrow) * GRIDW;
    }

    float m[8], l[8];
#pragma unroll
    for (int v = 0; v < 8; ++v) { m[v] = -3.0e38f; l[v] = 0.f; }
    v8f acc0 = {}, acc1 = {};

    for (int j0 = 0; j0 < KVLEN; j0 += 16) {
        // ---- S(16x16) = Qtile(16x32) x K_chunk^T(32x16) ----
        v8f s = {};
        const int kv = j0 + r;
        const float* kp = khat + ((size_t)(b * KVLEN + kv)) * DIMC + h * HD + kb0;
#pragma unroll
        for (int kk = 0; kk < 8; ++kk) {
            const v2f bf = *(const v2f*)(kp + kk * 4);   // B[k][n]=K[kv][k]
            s = wmma4(aq[kk], bf, s);
        }

        // ---- add rpe + mask (per-element in D layout) ----
        const float maskadd = (float)amask[b * KVLEN + kv] * -100.0f;
        const int x = kv & (GRIDW - 1), y = kv >> 6;
        float sv[8];
#pragma unroll
        for (int v = 0; v < 8; ++v)
            sv[v] = s[v] + rpex[(rpebase[v] + x) * NHEADS + h]
                         + rpey[(rpebase[v] + y) * NHEADS + h] + maskadd;

        // ---- online softmax (row = 16-lane group) ----
        float rescale[8];
#pragma unroll
        for (int v = 0; v < 8; ++v) {
            float rm = sv[v];
            rm = fmaxf(rm, __shfl_xor(rm, 1, 16));
            rm = fmaxf(rm, __shfl_xor(rm, 2, 16));
            rm = fmaxf(rm, __shfl_xor(rm, 4, 16));
            rm = fmaxf(rm, __shfl_xor(rm, 8, 16));
            const float nm = fmaxf(m[v], rm);
            const float sc = expf(m[v] - nm);
            const float p  = expf(sv[v] - nm);
            float rs = p;
            rs += __shfl_xor(rs, 1, 16);
            rs += __shfl_xor(rs, 2, 16);
            rs += __shfl_xor(rs, 4, 16);
            rs += __shfl_xor(rs, 8, 16);
            l[v] = l[v] * sc + rs;
            m[v] = nm;
            rescale[v] = sc;
            ldsP[((hi << 3) + v) * 16 + r] = p;   // D layout -> dense tile
        }
#pragma unroll
        for (int v = 0; v < 8; ++v) { acc0[v] *= rescale[v]; acc1[v] *= rescale[v]; }
        __syncthreads();

        // ---- ctx += P(16x16) x V_chunk(16x32) : two 16-wide output slabs ----
        const float* vp = vhat + ((size_t)(b * KVLEN + j0)) * DIMC + h * HD;
#pragma unroll
        for (int kk = 0; kk < 4; ++kk) {
            const v2f ap = *(const v2f*)(&ldsP[r * 16 + kk * 4 + kb0]); // A-frag of P
            const int krow = kk * 4 + kb0;
            v2f b0, b1;
            b0.x = vp[(size_t)krow * DIMC + r];
            b0.y = vp[(size_t)(krow + 1) * DIMC + r];
            b1.x = vp[(size_t)krow * DIMC + 16 + r];
            b1.y = vp[(size_t)(krow + 1) * DIMC + 16 + r];
            acc0 = wmma4(ap, b0, acc0);
            acc1 = wmma4(ap, b1, acc1);
        }
        __syncthreads();
    }

    // ---- normalize + store ctx (only here do we predicate) ----
#pragma unroll
    for (int v = 0; v < 8; ++v) {
        const int qrow = q0 + (hi << 3) + v;
        if (qrow < QLEN) {
            const float invl = 1.f / l[v];
            float* cp = ctx + ((size_t)(b * QLEN + qrow)) * DIMC + h * HD;
            cp[r]      = acc0[v] * invl;
            cp[16 + r] = acc1[v] * invl;
        }
    }
}

// ---------------------------------------------------------------------------
extern "C" void kernel_launch(void* const* d_in, const int* in_sizes, int n_in,
                              void* d_out, int out_size, void* d_ws, size_t ws_size,
                              hipStream_t stream) {
    (void)in_sizes; (void)n_in; (void)out_size; (void)ws_size;
    const float* hs   = (const float*)d_in[0];
    const float* refp = (const float*)d_in[1];
    const float* kvs  = (const float*)d_in[2];
    /* d_in[3] = spatial_shapes (64,64) — compile-time constants here */
    const int*   amk  = (const int*)d_in[4];
    const float* m1w1 = (const float*)d_in[5];
    const float* m1b1 = (const float*)d_in[6];
    const float* m1w2 = (const float*)d_in[7];
    const float* m2w1 = (const float*)d_in[8];
    const float* m2b1 = (const float*)d_in[9];
    const float* m2w2 = (const float*)d_in[10];
    const float* qw = (const float*)d_in[11]; const float* qb = (const float*)d_in[12];
    const float* kw = (const float*)d_in[13]; const float* kb = (const float*)d_in[14];
    const float* vw = (const float*)d_in[15]; const float* vb = (const float*)d_in[16];
    const float* ow = (const float*)d_in[17]; const float* ob = (const float*)d_in[18];

    float* ws   = (float*)d_ws;
    float* qhat = ws; ws += (size_t)BATCH * QLEN  * DIMC;
    float* khat = ws; ws += (size_t)BATCH * KVLEN * DIMC;
    float* vhat = ws; ws += (size_t)BATCH * KVLEN * DIMC;
    float* ctx  = ws; ws += (size_t)BATCH * QLEN  * DIMC;
    float* rpex = ws; ws += (size_t)BATCH * QLEN * GRIDW * NHEADS;
    float* rpey = ws; ws += (size_t)BATCH * QLEN * GRIDW * NHEADS;

    const int   MQ    = BATCH * QLEN;    // 1800
    const int   MKV   = BATCH * KVLEN;   // 8192
    const float scale = 0.17677669529663687f;  // (DIM/HEADS)^-0.5 = 32^-0.5

    const dim3 wave(32);
    proj_gemm64<<<dim3((MQ  + 15) / 16, DIMC / 64), wave, 0, stream>>>(hs,  qw, qb, qhat, MQ,  scale);
    proj_gemm64<<<dim3((MKV + 15) / 16, DIMC / 64), wave, 0, stream>>>(kvs, kw, kb, khat, MKV, 1.0f);
    proj_gemm64<<<dim3((MKV + 15) / 16, DIMC / 64), wave, 0, stream>>>(kvs, vw, vb, vhat, MKV, 1.0f);
    rpe_kernel<<<dim3(MQ), dim3(128), 0, stream>>>(refp, m1w1, m1b1, m1w2, m2w1, m2b1, m2w2, rpex, rpey);
    attn_kernel<<<dim3((QLEN + 15) / 16, NHEADS, BATCH), wave, 0, stream>>>(qhat, khat, vhat, rpex, rpey, amk, ctx);
    proj_gemm64<<<dim3((MQ + 15) / 16, DIMC / 64), wave, 0, stream>>>(ctx, ow, ob, (float*)d_out, MQ, 1.0f);
}